// AttentionLayer_79766132622013
// MI455X (gfx1250) — compile-verified
//
#include <hip/hip_runtime.h>
#include <math.h>
#include <stdint.h>

// Problem constants (from reference): B=16, T=12 -> BT=192, N=1024, D=128, H=8, hd=16
#define SEQ      1024
#define DIM      128
#define NHEAD    8
#define HD       16
#define BT       192
#define TROWS    64
#define NTHREADS 512
#define NTILES   (SEQ / TROWS)   // 16

typedef float v2f __attribute__((ext_vector_type(2)));
typedef float v8f __attribute__((ext_vector_type(8)));

#if defined(__has_builtin)
#if __has_builtin(__builtin_amdgcn_tensor_load_to_lds) && \
    __has_builtin(__builtin_amdgcn_s_wait_tensorcnt)
#define HAVE_TDM 1
#endif
#endif
#ifndef HAVE_TDM
#define HAVE_TDM 0
#endif

// D = A(16x4) * B(4x16) + C(16x16), fp32, wave32.
static __device__ __forceinline__ v8f wmma4(v2f a, v2f b, v8f c) {
  return __builtin_amdgcn_wmma_f32_16x16x4_f32(false, a, false, b, (short)0, c,
                                               false, false);
}

// Sum across each 16-lane half of a wave32 (xor masks 1,2,4,8 stay in-half).
static __device__ __forceinline__ float red16(float v) {
  v += __shfl_xor(v, 1, 32);
  v += __shfl_xor(v, 2, 32);
  v += __shfl_xor(v, 4, 32);
  v += __shfl_xor(v, 8, 32);
  return v;
}

#if HAVE_TDM
typedef unsigned int u32x4 __attribute__((ext_vector_type(4)));
typedef int          i32x4 __attribute__((ext_vector_type(4)));
typedef int          i32x8 __attribute__((ext_vector_type(8)));

// Tensor Data Mover: load a 2D tile of f32 (tile_d1 rows x tile_d0 elements,
// source row stride d0_stride elements) into contiguous LDS at ldst.
// D# built per CDNA5 ISA ch.8: group0 = {count, lds_addr, global_addr, type=2},
// group1 = {data_size=4B, tensor_dim0/1, tile_dim0/1, dim0_stride}, groups 2/3
// zero (higher dims unused). tile_d1 == 0 -> 1D copy of tile_d0 elements.
// This toolchain's builtin takes 6 args (g0, g1, g2, g3, extra, cpol); the
// trailing vector corresponds to descriptor state we leave zero.
static __device__ __forceinline__ void tdm_load_2d(const void* gsrc, void* ldst,
                                                   unsigned tile_d0,
                                                   unsigned tile_d1,
                                                   unsigned tensor_d0,
                                                   unsigned tensor_d1,
                                                   unsigned d0_stride) {
  uint64_t ga = (uint64_t)(uintptr_t)gsrc;
  u32x4 g0;
  g0[0] = 1u;                               // count=1, user descriptor
  g0[1] = (unsigned)(uintptr_t)ldst;        // LDS byte address = flat addr[31:0]
  g0[2] = (unsigned)ga;                     // global_addr[31:0]
  g0[3] = (unsigned)(ga >> 32) | (2u << 30);// global_addr[56:32] | type=2
  i32x8 g1;
  g1[0] = (int)(2u << 16);                  // data_size = 4 bytes
  g1[1] = (int)(tensor_d0 << 16);           // tensor_dim0[15:0]
  g1[2] = (int)((tensor_d0 >> 16) | (tensor_d1 << 16));
  g1[3] = (int)((tensor_d1 >> 16) | (tile_d0 << 16));
  g1[4] = (int)(tile_d1 & 0xFFFFu);         // tile_dim1 (0 = 1D), tile_dim2=0
  g1[5] = (int)d0_stride;                   // tensor_dim0_stride[31:0]
  g1[6] = 0;
  g1[7] = 0;
  i32x4 z4 = {0, 0, 0, 0};
  i32x8 z8 = {0, 0, 0, 0, 0, 0, 0, 0};
  __builtin_amdgcn_tensor_load_to_lds(g0, g1, z4, z4, z8, 0);
}

static __device__ __forceinline__ void tdm_load_1d(const void* gsrc, void* ldst,
                                                   unsigned nelem) {
  tdm_load_2d(gsrc, ldst, nelem, 0, nelem, 1, nelem);
}
#endif  // HAVE_TDM

// ---------------------------------------------------------------------------
// Kernel 1: per (b) compute partial kvs[b,h] = sum_n khat[n] (x) v[n]  (16x16)
// and partial ksum[b,h] = sum_n khat[n], via k = x@Wk, v = x@Wv.
// One block per b, 16 waves. TDM double-buffers the x tiles (DMA overlaps the
// WMMA compute); Wk|Wv panel staged once by a strided 2D TDM.
// ---------------------------------------------------------------------------
__global__ void __launch_bounds__(NTHREADS)
kv_stats_kernel(const float* __restrict__ x, const float* __restrict__ Wqkv,
                float* __restrict__ ws_kvs, float* __restrict__ ws_ksum) {
  extern __shared__ float smem[];
  float* w_lds  = smem;                   // 128*256 : W_qkv cols 128..383 = [Wk|Wv]
  float* x0_lds = w_lds + DIM * 256;      // 64*128 (ping)
  float* x1_lds = x0_lds + TROWS * DIM;   // 64*128 (pong)
  float* kv_lds = x1_lds + TROWS * DIM;   // 64*256 : [khat | v]

  const int b    = blockIdx.x;
  const int tid  = threadIdx.x;
  const int wave = tid >> 5;
  const int lane = tid & 31;
  const int half = lane >> 4;
  const int l    = lane & 15;

  const float* xb = x + (size_t)b * SEQ * DIM;

#if HAVE_TDM
  if (wave == 0) {
    // [Wk|Wv]: 128 rows x 256 cols out of the 384-wide W_qkv, packed 256-wide.
    tdm_load_2d(Wqkv + 128, w_lds, 256, 128, 384, 128, 384);
    tdm_load_1d(xb, x0_lds, TROWS * DIM);  // tile 0
  }
#else
  for (int e = tid; e < DIM * 256 / 4; e += NTHREADS) {
    int r  = e >> 6;
    int c4 = (e & 63) << 2;
    *(float4*)(w_lds + r * 256 + c4) =
        *(const float4*)(Wqkv + r * 384 + 128 + c4);
  }
#endif

  const int kh    = wave & 7;   // head this wave reduces
  const int rhalf = wave >> 3;  // which 32-row half of the tile
  const int rb    = wave & 3;   // GEMM role: row block (16 rows)
  const int qt    = wave >> 2;  // GEMM role: 64-col quarter (0,1 = k; 2,3 = v)
  const int row0  = rb * 16;
  const int c0    = qt * 64;

  v8f   kvacc = {0.f, 0.f, 0.f, 0.f, 0.f, 0.f, 0.f, 0.f};
  float ksum  = 0.f;

  for (int it = 0; it < NTILES; ++it) {
    __syncthreads();  // prior iteration's x / kv_lds readers are done
    float* x_lds = (it & 1) ? x1_lds : x0_lds;
#if HAVE_TDM
    if (wave == 0) {
      if (it + 1 < NTILES) {
        float* nbuf = ((it + 1) & 1) ? x1_lds : x0_lds;
        tdm_load_1d(xb + (size_t)(it + 1) * TROWS * DIM, nbuf, TROWS * DIM);
        __builtin_amdgcn_s_wait_tensorcnt(1);  // tile it (and W) complete
      } else {
        __builtin_amdgcn_s_wait_tensorcnt(0);
      }
    }
#else
    {
      const float* xt = xb + (size_t)it * TROWS * DIM;
      for (int e = tid; e < TROWS * DIM / 4; e += NTHREADS)
        *(float4*)(x_lds + (e << 2)) = *(const float4*)(xt + (e << 2));
      if (it + 1 < NTILES)
        __builtin_prefetch(xt + TROWS * DIM + tid * 16, 0, 0);
    }
#endif
    __syncthreads();

    // ---- [k|v] GEMM: each wave computes a 16x64 strip = 4 wmma tiles ----
    v8f acc[4];
#pragma unroll
    for (int t = 0; t < 4; ++t)
      acc[t] = (v8f){0.f, 0.f, 0.f, 0.f, 0.f, 0.f, 0.f, 0.f};
#pragma unroll 4
    for (int k0 = 0; k0 < DIM; k0 += 4) {
      // A[m][k]: m = lane%16 (x row), k = 2*half + {0,1}
      v2f a = *(const v2f*)(x_lds + (row0 + l) * DIM + k0 + 2 * half);
      const float* wr = w_lds + (k0 + 2 * half) * 256 + c0 + l;
#pragma unroll
      for (int t = 0; t < 4; ++t) {
        v2f bb = {wr[t * 16], wr[t * 16 + 256]};  // B[k][n], rows 256 apart
        acc[t] = wmma4(a, bb, acc[t]);
      }
    }

    // ---- L2-normalize k rows (quarters 0,1 hold k; each tile = one head) ----
    if (qt < 2) {
#pragma unroll
      for (int t = 0; t < 4; ++t) {
#pragma unroll
        for (int i = 0; i < 8; ++i) {
          float ss = red16(acc[t][i] * acc[t][i]);
          float sc = 1.f / fmaxf(sqrtf(ss), 1e-12f);
          acc[t][i] *= sc;
        }
      }
    }

    // ---- stash khat / v tiles into LDS (C-layout: row=i+8*half, col=lane%16)
#pragma unroll
    for (int t = 0; t < 4; ++t) {
      float* dst = kv_lds + c0 + t * 16 + l;
#pragma unroll
      for (int i = 0; i < 8; ++i)
        dst[(row0 + i + 8 * half) * 256] = acc[t][i];
    }
    __syncthreads();

    // ---- kvs += Khat^T * V for head kh over this wave's 32 rows ----
    // A[m][k] = khat[j0+k][16*kh+m]; B[k][d] = v[j0+k][16*kh+d]
#pragma unroll
    for (int js = 0; js < 8; ++js) {
      int j0 = rhalf * 32 + js * 4 + 2 * half;
      const float* kr = kv_lds + (size_t)j0 * 256 + kh * 16 + l;
      v2f a  = {kr[0], kr[256]};          // khat rows j0, j0+1
      v2f bb = {kr[128], kr[256 + 128]};  // v rows j0, j0+1
      ksum += a.x + a.y;                  // partial column sum of khat
      kvacc = wmma4(a, bb, kvacc);
    }
  }

  // ---- write partial stats (combined in kernel 2) ----
  float* kdst = ws_kvs + (((size_t)b * NHEAD + kh) * 2 + rhalf) * 256;
#pragma unroll
  for (int i = 0; i < 8; ++i)
    kdst[(i + 8 * half) * 16 + l] = kvacc[i];
  ws_ksum[(((size_t)b * NHEAD + kh) * 2 + rhalf) * 32 + lane] = ksum;
}

// ---------------------------------------------------------------------------
// Kernel 2: per (b, 64-row tile): q = x@Wq (normalize), v = x@Wv,
// attn = qhat@kvs + N*v, out_pre = attn / (qhat.ksum + N),
// out = out_pre @ W_out + b_out.  Fully fused; weights and x tile staged by TDM.
// ---------------------------------------------------------------------------
__global__ void __launch_bounds__(NTHREADS)
attn_out_kernel(const float* __restrict__ x, const float* __restrict__ Wqkv,
                const float* __restrict__ Wout, const float* __restrict__ bout,
                const float* __restrict__ ws_kvs,
                const float* __restrict__ ws_ksum, float* __restrict__ out) {
  extern __shared__ float smem[];
  float* wq_lds   = smem;                   // 128*128 (W_qkv cols 0..127)
  float* wv_lds   = wq_lds + DIM * DIM;     // 128*128 (W_qkv cols 256..383)
  float* wo_lds   = wv_lds + DIM * DIM;     // 128*128
  float* xq_lds   = wo_lds + DIM * DIM;     // 64*128 (x, later qhat)
  float* op_lds   = xq_lds + TROWS * DIM;   // 64*128 pre-projection
  float* kvs_lds  = op_lds + TROWS * DIM;   // 8*256
  float* ksum_lds = kvs_lds + NHEAD * 256;  // 128
  float* bias_lds = ksum_lds + DIM;         // 128

  const int b    = blockIdx.y;
  const int nt   = blockIdx.x;
  const int tid  = threadIdx.x;
  const int wave = tid >> 5;
  const int lane = tid & 31;
  const int half = lane >> 4;
  const int l    = lane & 15;

  const float* xt = x + ((size_t)b * SEQ + (size_t)nt * TROWS) * DIM;

  // ---- stage weights / x tile (TDM) and stats (VALU combine) ----
#if HAVE_TDM
  if (wave == 0) {
    tdm_load_2d(Wqkv, wq_lds, 128, 128, 384, 128, 384);
    tdm_load_2d(Wqkv + 256, wv_lds, 128, 128, 384, 128, 384);
    tdm_load_1d(Wout, wo_lds, DIM * DIM);
    tdm_load_1d(xt, xq_lds, TROWS * DIM);
  }
#else
  for (int e = tid; e < DIM * DIM / 4; e += NTHREADS) {
    int r = e >> 5;            // 32 float4 per 128-wide row
    int c = (e & 31) << 2;
    *(float4*)(wq_lds + r * 128 + c) = *(const float4*)(Wqkv + r * 384 + c);
    *(float4*)(wv_lds + r * 128 + c) =
        *(const float4*)(Wqkv + r * 384 + 256 + c);
  }
  for (int e = tid; e < DIM * DIM / 4; e += NTHREADS)
    *(float4*)(wo_lds + (e << 2)) = *(const float4*)(Wout + (e << 2));
  for (int e = tid; e < TROWS * DIM / 4; e += NTHREADS)
    *(float4*)(xq_lds + (e << 2)) = *(const float4*)(xt + (e << 2));
#endif
  for (int i = tid; i < NHEAD * 256; i += NTHREADS) {
    size_t base = (((size_t)b * NHEAD + (i >> 8)) * 2) * 256;
    int md = i & 255;
    kvs_lds[i] = ws_kvs[base + md] + ws_kvs[base + 256 + md];
  }
  if (tid < DIM) {
    size_t base = (((size_t)b * NHEAD + (tid >> 4)) * 2) * 32;
    int m = tid & 15;
    ksum_lds[tid] = ws_ksum[base + m] + ws_ksum[base + m + 16] +
                    ws_ksum[base + 32 + m] + ws_ksum[base + 48 + m];
    bias_lds[tid] = bout[tid];
  }
#if HAVE_TDM
  if (wave == 0) __builtin_amdgcn_s_wait_tensorcnt(0);
#endif
  __syncthreads();

  // ---- q/v GEMM: wave -> row block rb, heads h0, h0+1 (q and v tiles) ----
  const int rb   = wave & 3;
  const int h0   = (wave >> 2) * 2;
  const int row0 = rb * 16;
  v8f q0  = {0.f, 0.f, 0.f, 0.f, 0.f, 0.f, 0.f, 0.f};
  v8f q1  = q0, v0t = q0, v1t = q0;
#pragma unroll 4
  for (int k0 = 0; k0 < DIM; k0 += 4) {
    v2f a = *(const v2f*)(xq_lds + (row0 + l) * DIM + k0 + 2 * half);
    const float* wrq = wq_lds + (k0 + 2 * half) * DIM + 16 * h0 + l;
    const float* wrv = wv_lds + (k0 + 2 * half) * DIM + 16 * h0 + l;
    v2f bq0 = {wrq[0], wrq[DIM]};
    v2f bq1 = {wrq[16], wrq[16 + DIM]};
    v2f bv0 = {wrv[0], wrv[DIM]};
    v2f bv1 = {wrv[16], wrv[16 + DIM]};
    q0  = wmma4(a, bq0, q0);
    q1  = wmma4(a, bq1, q1);
    v0t = wmma4(a, bv0, v0t);
    v1t = wmma4(a, bv1, v1t);
  }

  // ---- L2-normalize q (per head = per tile; rows across 16-lane halves) ----
#pragma unroll
  for (int i = 0; i < 8; ++i) {
    float s0 = red16(q0[i] * q0[i]);
    q0[i] *= 1.f / fmaxf(sqrtf(s0), 1e-12f);
    float s1 = red16(q1[i] * q1[i]);
    q1[i] *= 1.f / fmaxf(sqrtf(s1), 1e-12f);
  }

  __syncthreads();  // everyone finished reading x from xq_lds
#pragma unroll
  for (int i = 0; i < 8; ++i) {
    xq_lds[(row0 + i + 8 * half) * DIM + 16 * h0 + l]       = q0[i];
    xq_lds[(row0 + i + 8 * half) * DIM + 16 * (h0 + 1) + l] = q1[i];
  }
  __syncthreads();

  // ---- attn = qhat @ kvs + N*v ; out_pre = attn / (qhat.ksum + N) ----
#pragma unroll
  for (int hh = 0; hh < 2; ++hh) {
    const int h = h0 + hh;
    v8f vt = (hh == 0) ? v0t : v1t;
    v8f qh = (hh == 0) ? q0 : q1;
    v8f att;
#pragma unroll
    for (int i = 0; i < 8; ++i) att[i] = 1024.f * vt[i];  // N * v (C layout)
#pragma unroll
    for (int k0 = 0; k0 < HD; k0 += 4) {
      v2f a = *(const v2f*)(xq_lds + (row0 + l) * DIM + 16 * h + k0 + 2 * half);
      const float* kr = kvs_lds + h * 256 + (k0 + 2 * half) * 16 + l;
      v2f bb = {kr[0], kr[16]};
      att = wmma4(a, bb, att);
    }
    float ks = ksum_lds[h * 16 + l];
#pragma unroll
    for (int i = 0; i < 8; ++i) {
      float dot = red16(qh[i] * ks);       // qhat-row . ksum  (per row)
      float inv = 1.f / (dot + 1024.f);    // + N
      op_lds[(row0 + i + 8 * half) * DIM + 16 * h + l] = att[i] * inv;
    }
  }
  __syncthreads();

  // ---- output projection: out = op @ W_out + b_out ----
  const int cb = wave & 7;
  const int rA = (wave >> 3) * 16;  // row blocks 0/16 and 32/48 per wave
  const int rB = rA + 32;
  const int c0 = cb * 16;
  v8f oA, oB;
#pragma unroll
  for (int i = 0; i < 8; ++i) {
    oA[i] = bias_lds[c0 + l];
    oB[i] = oA[i];
  }
#pragma unroll 4
  for (int k0 = 0; k0 < DIM; k0 += 4) {
    const float* wr = wo_lds + (k0 + 2 * half) * DIM + c0 + l;
    v2f bb = {wr[0], wr[DIM]};
    v2f aA = *(const v2f*)(op_lds + (rA + l) * DIM + k0 + 2 * half);
    v2f aB = *(const v2f*)(op_lds + (rB + l) * DIM + k0 + 2 * half);
    oA = wmma4(aA, bb, oA);
    oB = wmma4(aB, bb, oB);
  }
  float* ob = out + ((size_t)b * SEQ + (size_t)nt * TROWS) * DIM;
#pragma unroll
  for (int i = 0; i < 8; ++i) {
    ob[(rA + i + 8 * half) * DIM + c0 + l] = oA[i];
    ob[(rB + i + 8 * half) * DIM + c0 + l] = oB[i];
  }
}

extern "C" void kernel_launch(void* const* d_in, const int* in_sizes, int n_in,
                              void* d_out, int out_size, void* d_ws,
                              size_t ws_size, hipStream_t stream) {
  (void)in_sizes; (void)n_in; (void)out_size; (void)ws_size;
  const float* x    = (const float*)d_in[0];
  const float* Wqkv = (const float*)d_in[1];
  const float* Wout = (const float*)d_in[2];
  const float* bout = (const float*)d_in[3];
  float* out = (float*)d_out;

  float* ws_kvs  = (float*)d_ws;                           // 192*8*2*256 f32
  float* ws_ksum = ws_kvs + (size_t)BT * NHEAD * 2 * 256;  // 192*8*2*32  f32

  const size_t shmem1 =
      (size_t)(DIM * 256 + 2 * TROWS * DIM + TROWS * 256) * sizeof(float);
  const size_t shmem2 =
      (size_t)(3 * DIM * DIM + 2 * TROWS * DIM + NHEAD * 256 + DIM + DIM) *
      sizeof(float);

  (void)hipFuncSetAttribute((const void*)kv_stats_kernel,
                            hipFuncAttributeMaxDynamicSharedMemorySize,
                            (int)shmem1);
  (void)hipFuncSetAttribute((const void*)attn_out_kernel,
                            hipFuncAttributeMaxDynamicSharedMemorySize,
                            (int)shmem2);

  kv_stats_kernel<<<dim3(BT), dim3(NTHREADS), shmem1, stream>>>(x, Wqkv, ws_kvs,
                                                                ws_ksum);
  attn_out_kernel<<<dim3(NTILES, BT), dim3(NTHREADS), shmem2, stream>>>(
      x, Wqkv, Wout, bout, ws_kvs, ws_ksum, out);
}